// GraphConvLayer_71219147702937
// MI455X (gfx1250) — compile-verified
//
#include <hip/hip_runtime.h>
#include <cmath>
#include <stdint.h>

#define B_ 4
#define N_ 8192
#define D_ 64

typedef __attribute__((ext_vector_type(2))) float v2f;
typedef __attribute__((ext_vector_type(8))) float v8f;

__device__ __forceinline__ v8f wmma4(v2f a, v2f b, v8f c) {
  // D = A(16x4 f32) * B(4x16 f32) + C(16x16 f32)
  return __builtin_amdgcn_wmma_f32_16x16x4_f32(false, a, false, b, (short)0, c,
                                               false, false);
}

#if defined(__gfx1250__)
#define USE_ASYNC_LDS 1
// vdst = per-lane LDS byte address (low 32 bits of flat shared pointer),
// vaddr = 64-bit global address, no SADDR.
__device__ __forceinline__ void async_ld128(const void* g, void* l) {
  asm volatile("global_load_async_to_lds_b128 %0, %1, off"
               :
               : "v"((unsigned)(uintptr_t)l), "v"(g)
               : "memory");
}
__device__ __forceinline__ void wait_async0() {
  asm volatile("s_wait_asynccnt 0x0" ::: "memory");
}
#else
#define USE_ASYNC_LDS 0
#endif

#define KVS 68  // K/V tile row stride in floats: 16B-aligned, conflict-free S-frags

// ---------------------------------------------------------------------------
// Kernel 1: Q = x @ W1^T, K = x @ W2^T  (stored row-major in workspace)
// one wave per 16-row tile of the flattened (B*N, D) matrix
// ---------------------------------------------------------------------------
__global__ __launch_bounds__(128) void qk_kernel(
    const float* __restrict__ x, const float* __restrict__ W1,
    const float* __restrict__ W2, float* __restrict__ Qb,
    float* __restrict__ Kb) {
  const int lane = threadIdx.x & 31;
  const int wave = threadIdx.x >> 5;
  const int tile = blockIdx.x * 4 + wave;  // 16-row tile over B_*N_ rows
  const int row0 = tile * 16;
  const int m = lane & 15;
  const int h = lane >> 4;

  v2f xa[16];
  const float* xr = x + (size_t)(row0 + m) * D_;
#pragma unroll
  for (int k = 0; k < 16; ++k) {
    int d = 4 * k + 2 * h;
    xa[k] = (v2f){xr[d], xr[d + 1]};
  }

#pragma unroll
  for (int ec = 0; ec < 4; ++ec) {
    v8f q = {};
    v8f kk = {};
    const int e = ec * 16 + m;
#pragma unroll
    for (int k = 0; k < 16; ++k) {
      int d = 4 * k + 2 * h;
      v2f b1 = (v2f){W1[e * D_ + d], W1[e * D_ + d + 1]};
      v2f b2 = (v2f){W2[e * D_ + d], W2[e * D_ + d + 1]};
      q = wmma4(xa[k], b1, q);
      kk = wmma4(xa[k], b2, kk);
    }
#pragma unroll
    for (int v = 0; v < 8; ++v) {
      int r = row0 + (h ? 8 : 0) + v;
      Qb[(size_t)r * D_ + ec * 16 + m] = q[v];
      Kb[(size_t)r * D_ + ec * 16 + m] = kk[v];
    }
  }
}

// ---------------------------------------------------------------------------
// Kernel 2: fused flash-attention + GCN projection + ReLU
// block = 4 waves; each wave owns 32 query rows (two 16-row M-tiles) so every
// K/V LDS fragment feeds two WMMAs. Block streams K/V in 64-row LDS tiles.
// ---------------------------------------------------------------------------
__global__ __launch_bounds__(128) void attn_kernel(
    const float* __restrict__ x, const float* __restrict__ Qb,
    const float* __restrict__ Kb, const float* __restrict__ G,
    float* __restrict__ out) {
  __shared__ float Ks[64 * KVS];         // K tile
  __shared__ float Vs[64 * KVS];         // V tile (= x tile)
  __shared__ float Ps[4][2 * 16 * 18];   // per-wave P transpose buffer

  const int lane = threadIdx.x & 31;
  const int wave = threadIdx.x >> 5;
  const int b = blockIdx.x >> 6;         // 64 row-blocks (128 rows) per batch
  const int rb = blockIdx.x & 63;
  const int q0 = rb * 128 + wave * 32;   // this wave's first query row
  const int m = lane & 15;
  const int h = lane >> 4;
  const size_t base = (size_t)b * N_ * D_;

  // Q fragments held in registers: two 16x64 tiles -> 2x16 A-frags
  v2f qa[2][16];
#pragma unroll
  for (int rt = 0; rt < 2; ++rt) {
    const float* qr = Qb + base + (size_t)(q0 + rt * 16 + m) * D_;
#pragma unroll
    for (int k = 0; k < 16; ++k) {
      int d = 4 * k + 2 * h;
      qa[rt][k] = (v2f){qr[d], qr[d + 1]};
    }
  }

  v8f O[2][4];
  float mrow[2][8], srow[2][8];
#pragma unroll
  for (int rt = 0; rt < 2; ++rt) {
#pragma unroll
    for (int dc = 0; dc < 4; ++dc) O[rt][dc] = (v8f){};
#pragma unroll
    for (int v = 0; v < 8; ++v) {
      mrow[rt][v] = -3.0e38f;
      srow[rt][v] = 0.0f;
    }
  }

  for (int kv0 = 0; kv0 < N_; kv0 += 64) {
    const float* kg = Kb + base + (size_t)kv0 * D_;
    const float* vg = x + base + (size_t)kv0 * D_;
#if USE_ASYNC_LDS
#pragma unroll
    for (int i = 0; i < 8; ++i) {
      int idx = threadIdx.x + i * 128;  // float4 index 0..1023
      int r = idx >> 4;
      int c = (idx & 15) * 4;
      async_ld128(kg + r * 64 + c, Ks + r * KVS + c);
      async_ld128(vg + r * 64 + c, Vs + r * KVS + c);
    }
    wait_async0();
#else
#pragma unroll
    for (int i = 0; i < 8; ++i) {
      int idx = threadIdx.x + i * 128;
      int r = idx >> 4;
      int c = (idx & 15) * 4;
      float4 kf = *(const float4*)(kg + r * 64 + c);
      float4 vf = *(const float4*)(vg + r * 64 + c);
      float* kd = Ks + r * KVS + c;
      float* vd = Vs + r * KVS + c;
      kd[0] = kf.x; kd[1] = kf.y; kd[2] = kf.z; kd[3] = kf.w;
      vd[0] = vf.x; vd[1] = vf.y; vd[2] = vf.z; vd[3] = vf.w;
    }
#endif
    if (kv0 + 64 < N_) {  // next-tile prefetch -> global_prefetch_b8
      __builtin_prefetch(kg + 64 * D_ + threadIdx.x * 32, 0, 1);
      __builtin_prefetch(vg + 64 * D_ + threadIdx.x * 32, 0, 1);
    }
    __syncthreads();

#pragma unroll 1
    for (int st = 0; st < 4; ++st) {  // 16-column subtiles
      const int n0 = st * 16;
      v8f S[2];
      S[0] = (v8f){};
      S[1] = (v8f){};
#pragma unroll
      for (int k = 0; k < 16; ++k) {  // S = Q * K^T ; K frag reused 2x
        int d = 4 * k + 2 * h;
        const float* kr = Ks + (n0 + m) * KVS + d;
        v2f kb = (v2f){kr[0], kr[1]};
        S[0] = wmma4(qa[0][k], kb, S[0]);
        S[1] = wmma4(qa[1][k], kb, S[1]);
      }
      // online softmax (rows live in 16-lane halves; xor masks stay inside)
      float* pw = Ps[wave];
#pragma unroll
      for (int rt = 0; rt < 2; ++rt) {
#pragma unroll
        for (int v = 0; v < 8; ++v) {
          float t = S[rt][v];
          t = fmaxf(t, __shfl_xor(t, 1, 32));
          t = fmaxf(t, __shfl_xor(t, 2, 32));
          t = fmaxf(t, __shfl_xor(t, 4, 32));
          t = fmaxf(t, __shfl_xor(t, 8, 32));
          float mn = fmaxf(mrow[rt][v], t);
          float sc = __expf(mrow[rt][v] - mn);
          mrow[rt][v] = mn;
          float p = __expf(S[rt][v] - mn);
          float ps = p;
          ps += __shfl_xor(ps, 1, 32);
          ps += __shfl_xor(ps, 2, 32);
          ps += __shfl_xor(ps, 4, 32);
          ps += __shfl_xor(ps, 8, 32);
          srow[rt][v] = srow[rt][v] * sc + ps;
          O[rt][0][v] *= sc;
          O[rt][1][v] *= sc;
          O[rt][2][v] *= sc;
          O[rt][3][v] *= sc;
          // C-layout -> A-layout transpose of P via wave-private LDS
          pw[(rt * 16 + (h ? 8 : 0) + v) * 18 + m] = p;
        }
      }
      // same-wave DS ops are in-order: no barrier needed
#pragma unroll
      for (int nk = 0; nk < 4; ++nk) {  // O += P * V ; V frag reused 2x
        int nn = 4 * nk + 2 * h;
        v2f pa0 = (v2f){pw[m * 18 + nn], pw[m * 18 + nn + 1]};
        v2f pa1 = (v2f){pw[(16 + m) * 18 + nn], pw[(16 + m) * 18 + nn + 1]};
#pragma unroll
        for (int dc = 0; dc < 4; ++dc) {
          int col = dc * 16 + m;
          v2f vb =
              (v2f){Vs[(n0 + nn) * KVS + col], Vs[(n0 + nn + 1) * KVS + col]};
          O[0][dc] = wmma4(pa0, vb, O[0][dc]);
          O[1][dc] = wmma4(pa1, vb, O[1][dc]);
        }
      }
    }
    __syncthreads();
  }

  // normalize rows
#pragma unroll
  for (int rt = 0; rt < 2; ++rt)
#pragma unroll
    for (int v = 0; v < 8; ++v) {
      float inv = 1.0f / srow[rt][v];
      O[rt][0][v] *= inv;
      O[rt][1][v] *= inv;
      O[rt][2][v] *= inv;
      O[rt][3][v] *= inv;
    }

  // epilogue: out = relu(O @ G); transpose O through (now free) K/V tiles
#pragma unroll
  for (int rt = 0; rt < 2; ++rt) {
    float* ow = (rt == 0 ? Ks : Vs) + wave * (16 * KVS);
#pragma unroll
    for (int v = 0; v < 8; ++v) {
      int r = (h ? 8 : 0) + v;
      ow[r * KVS + 0 * 16 + m] = O[rt][0][v];
      ow[r * KVS + 1 * 16 + m] = O[rt][1][v];
      ow[r * KVS + 2 * 16 + m] = O[rt][2][v];
      ow[r * KVS + 3 * 16 + m] = O[rt][3][v];
    }
    v2f oa[16];
#pragma unroll
    for (int k = 0; k < 16; ++k) {
      int d = 4 * k + 2 * h;
      oa[k] = (v2f){ow[m * KVS + d], ow[m * KVS + d + 1]};
    }
    float* outp = out + base + (size_t)(q0 + rt * 16) * D_;
#pragma unroll
    for (int ec = 0; ec < 4; ++ec) {
      v8f acc = {};
#pragma unroll
      for (int k = 0; k < 16; ++k) {
        int d = 4 * k + 2 * h;
        // B[kk][e] = G[d][e], G row-major 64x64 (einsum 'bnd,de->bne')
        v2f gb = (v2f){G[d * D_ + ec * 16 + m], G[(d + 1) * D_ + ec * 16 + m]};
        acc = wmma4(oa[k], gb, acc);
      }
#pragma unroll
      for (int v = 0; v < 8; ++v) {
        int r = (h ? 8 : 0) + v;
        outp[(size_t)r * D_ + ec * 16 + m] = fmaxf(acc[v], 0.0f);
      }
    }
  }
}

// ---------------------------------------------------------------------------
extern "C" void kernel_launch(void* const* d_in, const int* in_sizes, int n_in,
                              void* d_out, int out_size, void* d_ws,
                              size_t ws_size, hipStream_t stream) {
  (void)in_sizes; (void)n_in; (void)out_size; (void)ws_size;
  const float* x  = (const float*)d_in[0];
  const float* W1 = (const float*)d_in[1];
  const float* W2 = (const float*)d_in[2];
  const float* G  = (const float*)d_in[3];
  float* out = (float*)d_out;
  float* Qb = (float*)d_ws;                   // 4*8192*64 f32 = 8.4 MB
  float* Kb = Qb + (size_t)B_ * N_ * D_;      // another 8.4 MB

  qk_kernel<<<(B_ * N_ / 16) / 4, 128, 0, stream>>>(x, W1, W2, Qb, Kb);
  attn_kernel<<<B_ * (N_ / 128), 128, 0, stream>>>(x, Qb, Kb, G, out);
}